// MinGRUcell_log_47605417509425
// MI455X (gfx1250) — compile-verified
//
#include <hip/hip_runtime.h>
#include <math.h>

// MinGRU log-space scan for MI455X (gfx1250, wave32).
// Phase 1: dual fp32 WMMA GEMM (k = x*Wz^T+bz, p = x*Wh^T+bh) fused with the
//          log-space elementwise math and the per-(b,t) cumsum over H.
// Phase 2: chunked (associative) cumlogsumexp over T.

typedef float v2f __attribute__((ext_vector_type(2)));
typedef float v8f __attribute__((ext_vector_type(8)));

constexpr int B_   = 8;
constexpr int T_   = 4096;
constexpr int IN_  = 512;   // K of the GEMM
constexpr int H_   = 512;   // N of the GEMM
constexpr int ROWS = 32;    // M rows per block (2 M-tiles per wave)
constexpr int NCH  = 32;    // chunks along T for the parallel scan
constexpr int CLEN = T_ / NCH;   // 128
constexpr int CHN  = B_ * H_;    // 4096 independent scan channels

// ---------- math helpers (fp32, numerically stable) ----------
__device__ __forceinline__ float softplus_f(float a) {
  // log(1 + exp(a)) = max(a,0) + log1p(exp(-|a|))
  return fmaxf(a, 0.0f) + log1pf(expf(-fabsf(a)));
}
__device__ __forceinline__ float log_g_f(float x) {
  // x>=0: log(x + 0.5) ; x<0: -softplus(-x)
  return (x >= 0.0f) ? logf(x + 0.5f) : -softplus_f(-x);
}
__device__ __forceinline__ float logaddexp_f(float a, float b) {
  float mx = fmaxf(a, b);
  float mn = fminf(a, b);
  return mx + log1pf(expf(mn - mx));   // finite inputs only
}

// ---------------------------------------------------------------------------
// Kernel 1: fused dual GEMM (fp32 WMMA) + elementwise + cumsum over H.
// Block: 256 threads (8 waves). Panel: 32 rows x 512 cols.
// Wave w covers cols [w*64, w*64+64), rows [0,32) as 2 M-tiles x 4 N-tiles.
// Writes: Aarr[b,t,h] = cumsum_h(log_coeffs[b,t,:])   (== a_star at t+1)
//         Darr[b,t,h] = (log_z + log_tilde_h) - Aarr  (== log_values - a_star)
// ---------------------------------------------------------------------------
__global__ __launch_bounds__(256)
void mingru_gemm_kernel(const float* __restrict__ x,
                        const float* __restrict__ Wz, const float* __restrict__ bz,
                        const float* __restrict__ Wh, const float* __restrict__ bh,
                        float* __restrict__ Aarr, float* __restrict__ Darr) {
  __shared__ float lds_c[ROWS][H_ + 1];   // log_coeffs -> in-place cumsum
  __shared__ float lds_v[ROWS][H_ + 1];   // log_values (z+tilde_h part)

  const int tid  = threadIdx.x;
  const int wave = tid >> 5;
  const int lane = tid & 31;
  const int rowBase = blockIdx.x * ROWS;      // multiple of 32; never straddles b
  const int n0   = wave * 64;
  const int lm   = lane & 15;
  const int kb   = (lane >> 4) * 2;           // K sub-offset for A/B fragments
  const int moff = (lane >> 4) * 8;           // C/D row offset for upper lanes

  v8f accZ[2][4];
  v8f accH[2][4];
#pragma unroll
  for (int mt = 0; mt < 2; ++mt)
#pragma unroll
    for (int nt = 0; nt < 4; ++nt) { accZ[mt][nt] = (v8f){}; accH[mt][nt] = (v8f){}; }

  const float* xr0 = x + (size_t)(rowBase + lm) * IN_;
  const float* xr1 = x + (size_t)(rowBase + 16 + lm) * IN_;

  for (int ko = 0; ko < IN_ / 4; ++ko) {
    const int k0 = ko * 4 + kb;
    // A fragments: 16x4 f32, lane m = lane&15, VGPR e -> K = kb + e
    v2f a0 = *(const v2f*)(xr0 + k0);
    v2f a1 = *(const v2f*)(xr1 + k0);
#pragma unroll
    for (int nt = 0; nt < 4; ++nt) {
      const int col = n0 + nt * 16 + lm;
      // B fragments: 4x16 f32, B[k][n] = W[n*IN + k]
      v2f bzf = *(const v2f*)(Wz + (size_t)col * IN_ + k0);
      v2f bhf = *(const v2f*)(Wh + (size_t)col * IN_ + k0);
      accZ[0][nt] = __builtin_amdgcn_wmma_f32_16x16x4_f32(false, a0, false, bzf,
                                                          (short)0, accZ[0][nt], false, false);
      accZ[1][nt] = __builtin_amdgcn_wmma_f32_16x16x4_f32(false, a1, false, bzf,
                                                          (short)0, accZ[1][nt], false, false);
      accH[0][nt] = __builtin_amdgcn_wmma_f32_16x16x4_f32(false, a0, false, bhf,
                                                          (short)0, accH[0][nt], false, false);
      accH[1][nt] = __builtin_amdgcn_wmma_f32_16x16x4_f32(false, a1, false, bhf,
                                                          (short)0, accH[1][nt], false, false);
    }
  }

  // Unpack C/D layout (VGPR j: lanes 0-15 -> M=j, lanes 16-31 -> M=j+8),
  // apply bias + log-space elementwise, stash to LDS.
#pragma unroll
  for (int mt = 0; mt < 2; ++mt)
#pragma unroll
    for (int nt = 0; nt < 4; ++nt) {
      const int col = n0 + nt * 16 + lm;
      const float bzv = bz[col];
      const float bhv = bh[col];
#pragma unroll
      for (int j = 0; j < 8; ++j) {
        const int row = mt * 16 + j + moff;
        const float kv = accZ[mt][nt][j] + bzv;   // pre-sigmoid gate
        const float pv = accH[mt][nt][j] + bhv;   // pre-activation hidden
        lds_c[row][col] = -softplus_f(kv);                      // log(1-sigmoid(k))
        lds_v[row][col] = -softplus_f(-kv) + log_g_f(pv);       // log_z + log_tilde_h
      }
    }
  __syncthreads();

  // Inclusive cumsum of log_coeffs along H (512) per row.
  // Each wave scans 4 rows: 16 serial elems/lane + shfl-up wave prefix.
#pragma unroll
  for (int rr = 0; rr < 4; ++rr) {
    const int row = wave * 4 + rr;
    float v[16];
    float run = 0.0f;
    const int base = lane * 16;
#pragma unroll
    for (int i = 0; i < 16; ++i) { run += lds_c[row][base + i]; v[i] = run; }
    const float tot = run;
    float sc = tot;
#pragma unroll
    for (int d = 1; d < 32; d <<= 1) {
      float t = __shfl_up(sc, d, 32);
      if (lane >= d) sc += t;
    }
    const float excl = sc - tot;
#pragma unroll
    for (int i = 0; i < 16; ++i) lds_c[row][base + i] = v[i] + excl;
  }
  __syncthreads();

  // Coalesced write-out of Aarr / Darr.
  const int b     = rowBase >> 12;          // T_ = 4096
  const int tbase = rowBase & (T_ - 1);
#pragma unroll 4
  for (int it = 0; it < (ROWS * H_) / 256; ++it) {
    const int idx  = it * 256 + tid;
    const int r    = idx >> 9;
    const int hcol = idx & (H_ - 1);
    const float cs = lds_c[r][hcol];
    const size_t g = ((size_t)b * T_ + (tbase + r)) * H_ + hcol;
    Aarr[g] = cs;
    Darr[g] = lds_v[r][hcol] - cs;
  }
}

// ---------------------------------------------------------------------------
// Kernel 2: per-(channel, chunk) logsumexp reduction over CLEN steps.
// ---------------------------------------------------------------------------
__global__ __launch_bounds__(256)
void mingru_chunk_reduce_kernel(const float* __restrict__ Darr,
                                float* __restrict__ totals) {
  const int idx = blockIdx.x * 256 + threadIdx.x;   // 0 .. CHN*NCH-1
  const int ch  = idx & (CHN - 1);
  const int c   = idx >> 12;                        // CHN = 4096
  const int b   = ch >> 9;
  const int h   = ch & (H_ - 1);
  const float* p = Darr + ((size_t)b * T_ + c * CLEN) * H_ + h;
  float acc = p[0];
  for (int i = 1; i < CLEN; ++i) acc = logaddexp_f(acc, p[(size_t)i * H_]);
  totals[(size_t)c * CHN + ch] = acc;
}

// ---------------------------------------------------------------------------
// Kernel 3: per-channel exclusive prefix over chunk totals, seeded with the
// padded t=0 term log_g(h0) (a_star=0 there).
// ---------------------------------------------------------------------------
__global__ __launch_bounds__(256)
void mingru_chunk_prefix_kernel(const float* __restrict__ h0,
                                const float* __restrict__ totals,
                                float* __restrict__ prefix) {
  const int ch = blockIdx.x * 256 + threadIdx.x;    // 0 .. CHN-1
  const int b  = ch >> 9;
  const int h  = ch & (H_ - 1);
  float run = log_g_f(h0[b * H_ + h]);
#pragma unroll
  for (int c = 0; c < NCH; ++c) {
    prefix[(size_t)c * CHN + ch] = run;             // lse over {t=0} U chunks < c
    run = logaddexp_f(run, totals[(size_t)c * CHN + ch]);
  }
}

// ---------------------------------------------------------------------------
// Kernel 4: final pass — running logaddexp within the chunk, emit
// h[b,t,h] = exp(a_star[t+1] + cumlse[t+1]); chunk 31 also writes h[:, -1].
// ---------------------------------------------------------------------------
__global__ __launch_bounds__(256)
void mingru_final_kernel(const float* __restrict__ Aarr,
                         const float* __restrict__ Darr,
                         const float* __restrict__ prefix,
                         float* __restrict__ out,
                         float* __restrict__ out_last) {
  const int idx = blockIdx.x * 256 + threadIdx.x;
  const int ch  = idx & (CHN - 1);
  const int c   = idx >> 12;
  const int b   = ch >> 9;
  const int h   = ch & (H_ - 1);
  float run = prefix[(size_t)c * CHN + ch];
  const size_t base = ((size_t)b * T_ + c * CLEN) * H_ + h;
  float last = 0.0f;
  for (int i = 0; i < CLEN; ++i) {
    const size_t g = base + (size_t)i * H_;
    run = logaddexp_f(run, Darr[g]);
    last = expf(Aarr[g] + run);
    out[g] = last;
  }
  if (c == NCH - 1) out_last[b * H_ + h] = last;
}

// ---------------------------------------------------------------------------
extern "C" void kernel_launch(void* const* d_in, const int* in_sizes, int n_in,
                              void* d_out, int out_size, void* d_ws, size_t ws_size,
                              hipStream_t stream) {
  (void)in_sizes; (void)n_in; (void)out_size; (void)ws_size;
  const float* x  = (const float*)d_in[0];   // [B,T,IN]
  const float* h0 = (const float*)d_in[1];   // [B,1,H]
  const float* Wz = (const float*)d_in[2];   // [H,IN]
  const float* bz = (const float*)d_in[3];   // [H]
  const float* Wh = (const float*)d_in[4];   // [H,IN]
  const float* bh = (const float*)d_in[5];   // [H]

  float* out      = (float*)d_out;                     // [B,T,H]
  float* out_last = out + (size_t)B_ * T_ * H_;        // [B,1,H]

  float* Aarr   = (float*)d_ws;                        // [B,T,H]  a_star(t+1)
  float* Darr   = Aarr + (size_t)B_ * T_ * H_;         // [B,T,H]  values - a_star
  float* totals = Darr + (size_t)B_ * T_ * H_;         // [NCH, CHN]
  float* prefix = totals + (size_t)NCH * CHN;          // [NCH, CHN]

  mingru_gemm_kernel<<<(B_ * T_) / ROWS, 256, 0, stream>>>(x, Wz, bz, Wh, bh, Aarr, Darr);
  mingru_chunk_reduce_kernel<<<(CHN * NCH) / 256, 256, 0, stream>>>(Darr, totals);
  mingru_chunk_prefix_kernel<<<CHN / 256, 256, 0, stream>>>(h0, totals, prefix);
  mingru_final_kernel<<<(CHN * NCH) / 256, 256, 0, stream>>>(Aarr, Darr, prefix, out, out_last);
}